// SAGEMalwareClassifier_24137716203811
// MI455X (gfx1250) — compile-verified
//
#include <hip/hip_runtime.h>

// Problem constants (match reference)
#define NN 100000
#define NE 1250000
#define DD 64
#define GG 128
#define CLF_IN 68   // H + 4 graph attrs
#define CLF_H 32    // H/2
#define BN_EPS 1e-5f

#define LDS_STRIDE 68   // 64 + 4 pad: stride ≡ 4 (mod 64 banks) -> conflict-free b64 reads

typedef float v2f __attribute__((ext_vector_type(2)));
typedef float v8f __attribute__((ext_vector_type(8)));

// ---------------------------------------------------------------- utilities
__global__ void zero_f32(float* __restrict__ p, int n) {
  int i = blockIdx.x * blockDim.x + threadIdx.x;
  int stride = gridDim.x * blockDim.x;
  for (; i < n; i += stride) p[i] = 0.0f;
}

// ---------------------------------------------------------------- degrees
__global__ void count_edges(const int* __restrict__ dst, float* __restrict__ cnt) {
  int e = blockIdx.x * blockDim.x + threadIdx.x;
  if (e < NE) atomicAdd(&cnt[dst[e]], 1.0f);
}

__global__ void invert_cnt(float* __restrict__ cnt) {
  int i = blockIdx.x * blockDim.x + threadIdx.x;
  if (i < NN) cnt[i] = 1.0f / fmaxf(cnt[i], 1.0f);
}

// ---------------------------------------------------------------- scatter-add of neighbor features
// one wave per edge; lane handles contiguous column pair -> one b64 gather per lane
__global__ __launch_bounds__(256)
void edge_aggregate(const float* __restrict__ h,
                    const int* __restrict__ src,
                    const int* __restrict__ dst,
                    float* __restrict__ agg) {
  int t = blockIdx.x * blockDim.x + threadIdx.x;
  int e = t >> 5;
  int lane = t & 31;
  if (e >= NE) return;
  int s = src[e];
  int d = dst[e];
  int col = lane * 2;
  const float2 v = *(const float2*)(h + (size_t)s * DD + col);
  float* a = agg + (size_t)d * DD + col;
  atomicAdd(a,     v.x);
  atomicAdd(a + 1, v.y);
}

// ---------------------------------------------------------------- fused SAGE layer:
//   out = ReLU( BN( (agg*inv_cnt) @ Wl + bl + h @ Wr ) )
// Weights staged TRANSPOSED in LDS once per block (ds_load_b64 B-fragments),
// one wave per 16-row node tile, 4 column tiles of 16x16 fp32 accumulators,
// V_WMMA_F32_16X16X4_F32 over K=64 (16 steps).
__global__ __launch_bounds__(256)
void sage_layer_wmma(const float* __restrict__ h_in,
                     const float* __restrict__ agg,
                     const float* __restrict__ inv_cnt,
                     const float* __restrict__ Wl,   // [64,64] row-major (in,out)
                     const float* __restrict__ bl,   // [64]
                     const float* __restrict__ Wr,   // [64,64]
                     const float* __restrict__ gbn,  // [64]
                     const float* __restrict__ bebn, // [64]
                     const float* __restrict__ rm,   // [64]
                     const float* __restrict__ rv,   // [64]
                     float* __restrict__ h_out) {
  __shared__ float lW[2 * DD * LDS_STRIDE];   // [path][n][k] transposed, padded

  // cooperative transpose-stage of Wl, Wr (coalesced global reads)
  for (int idx = threadIdx.x; idx < DD * DD; idx += 256) {
    int k = idx >> 6;          // input-dim index (row of W)
    int n = idx & 63;          // output-dim index (col of W)
    lW[n * LDS_STRIDE + k]                    = Wl[idx];
    lW[DD * LDS_STRIDE + n * LDS_STRIDE + k]  = Wr[idx];
  }
  __syncthreads();

  const int lane = threadIdx.x & 31;
  const int wave = threadIdx.x >> 5;
  const int tile = blockIdx.x * 8 + wave;       // 16-row tile index
  const int row0 = tile * 16;
  if (row0 >= NN) return;                        // wave-uniform: EXEC stays all-1s

  const int m    = lane & 15;                    // A-row / B-col within tile
  const int half = lane >> 4;                    // which K pair this lane holds

  const int arow = row0 + m;
  const float ic = inv_cnt[arow];

  v8f c0 = {}, c1 = {}, c2 = {}, c3 = {};

  const float* __restrict__ hrow = h_in + (size_t)arow * DD;
  const float* __restrict__ grow = agg  + (size_t)arow * DD;

  const float* lWl = lW;
  const float* lWr = lW + DD * LDS_STRIDE;

  #pragma unroll
  for (int k = 0; k < DD; k += 4) {
    const int kk = k + half * 2;                 // this lane's K indices: kk, kk+1 (even)
    const float2 ga = *(const float2*)(grow + kk);   // global_load_b64
    const float2 ha = *(const float2*)(hrow + kk);   // global_load_b64
    v2f a_agg, a_self;
    a_agg.x  = ga.x * ic;
    a_agg.y  = ga.y * ic;
    a_self.x = ha.x;
    a_self.y = ha.y;

    #pragma unroll
    for (int ct = 0; ct < 4; ++ct) {
      const int n = ct * 16 + m;
      const float2 fl = *(const float2*)(lWl + n * LDS_STRIDE + kk);  // ds_load_b64
      const float2 fr = *(const float2*)(lWr + n * LDS_STRIDE + kk);  // ds_load_b64
      v2f bL, bR;
      bL.x = fl.x; bL.y = fl.y;
      bR.x = fr.x; bR.y = fr.y;
      v8f& c = (ct == 0) ? c0 : (ct == 1) ? c1 : (ct == 2) ? c2 : c3;
      c = __builtin_amdgcn_wmma_f32_16x16x4_f32(false, a_agg,  false, bL, (short)0, c, false, false);
      c = __builtin_amdgcn_wmma_f32_16x16x4_f32(false, a_self, false, bR, (short)0, c, false, false);
    }
  }

  // epilogue: +bias, BatchNorm (eval), ReLU, store.
  // C/D layout: VGPR v -> row = row0 + v + half*8, col = ct*16 + (lane&15)
  v8f acc[4] = {c0, c1, c2, c3};
  #pragma unroll
  for (int ct = 0; ct < 4; ++ct) {
    const int n = ct * 16 + m;
    const float sc  = gbn[n] * rsqrtf(rv[n] + BN_EPS);
    const float bln = bl[n];
    const float rmn = rm[n];
    const float ben = bebn[n];
    #pragma unroll
    for (int v = 0; v < 8; ++v) {
      const int row = row0 + v + half * 8;
      float val = acc[ct][v];
      val = (val + bln - rmn) * sc + ben;
      h_out[(size_t)row * DD + n] = fmaxf(val, 0.0f);
    }
  }
}

// ---------------------------------------------------------------- global mean pool
// batch is sorted: each wave takes a contiguous run of 32 nodes and accumulates
// locally in registers, flushing to global atomics only on graph-id change.
#define POOL_CHUNK 32
__global__ __launch_bounds__(256)
void pool_nodes(const float* __restrict__ h,
                const int* __restrict__ batch,
                float* __restrict__ gsum,   // [G,64]
                float* __restrict__ gcnt) { // [G]
  int t = blockIdx.x * blockDim.x + threadIdx.x;
  int w = t >> 5;
  int lane = t & 31;
  const int nWaves = (NN + POOL_CHUNK - 1) / POOL_CHUNK;
  if (w >= nWaves) return;
  int n0 = w * POOL_CHUNK;
  int n1 = min(n0 + POOL_CHUNK, NN);
  int col = lane * 2;

  float a0 = 0.0f, a1 = 0.0f, c = 0.0f;
  int cur = batch[n0];
  for (int n = n0; n < n1; ++n) {
    int g = batch[n];                 // warp-uniform
    if (g != cur) {                   // uniform branch
      atomicAdd(&gsum[cur * DD + col],     a0);
      atomicAdd(&gsum[cur * DD + col + 1], a1);
      if (lane == 0) atomicAdd(&gcnt[cur], c);
      a0 = a1 = c = 0.0f;
      cur = g;
    }
    const float2 v = *(const float2*)(h + (size_t)n * DD + col);
    a0 += v.x;
    a1 += v.y;
    c  += 1.0f;
  }
  atomicAdd(&gsum[cur * DD + col],     a0);
  atomicAdd(&gsum[cur * DD + col + 1], a1);
  if (lane == 0) atomicAdd(&gcnt[cur], c);
}

// ---------------------------------------------------------------- classifier head (tiny)
__global__ __launch_bounds__(128)
void classifier(const float* __restrict__ gsum,
                const float* __restrict__ gcnt,
                const float* __restrict__ gattr, // [G,4]
                const float* __restrict__ W1,    // [68,32]
                const float* __restrict__ b1,    // [32]
                const float* __restrict__ W2,    // [32,2]
                const float* __restrict__ b2,    // [2]
                float* __restrict__ out) {       // [G,2]
  int g = blockIdx.x * blockDim.x + threadIdx.x;
  if (g >= GG) return;
  float inv = 1.0f / fmaxf(gcnt[g], 1.0f);
  float feat[CLF_IN];
  #pragma unroll
  for (int i = 0; i < DD; ++i) feat[i] = gsum[g * DD + i] * inv;
  #pragma unroll
  for (int i = 0; i < 4; ++i)  feat[DD + i] = gattr[g * 4 + i];

  float hid[CLF_H];
  #pragma unroll
  for (int j = 0; j < CLF_H; ++j) {
    float s = b1[j];
    for (int i = 0; i < CLF_IN; ++i) s += feat[i] * W1[i * CLF_H + j];
    hid[j] = fmaxf(s, 0.0f);
  }
  float o0 = b2[0], o1 = b2[1];
  #pragma unroll
  for (int j = 0; j < CLF_H; ++j) {
    o0 += hid[j] * W2[j * 2 + 0];
    o1 += hid[j] * W2[j * 2 + 1];
  }
  out[g * 2 + 0] = o0;
  out[g * 2 + 1] = o1;
}

// ---------------------------------------------------------------- launch
extern "C" void kernel_launch(void* const* d_in, const int* in_sizes, int n_in,
                              void* d_out, int out_size, void* d_ws, size_t ws_size,
                              hipStream_t stream) {
  (void)in_sizes; (void)n_in; (void)out_size; (void)ws_size;

  const float* x     = (const float*)d_in[0];
  const int*   eidx  = (const int*)d_in[1];      // [2,E]
  const int*   batch = (const int*)d_in[2];      // [N]
  const float* gattr = (const float*)d_in[3];    // [G,4]

  // per-layer params start at index 4, 7 per layer: Wl, bl, Wr, g, be, rm, rv
  const float* P[3][7];
  for (int l = 0; l < 3; ++l)
    for (int j = 0; j < 7; ++j)
      P[l][j] = (const float*)d_in[4 + l * 7 + j];

  const float* W1 = (const float*)d_in[25];
  const float* b1 = (const float*)d_in[26];
  const float* W2 = (const float*)d_in[27];
  const float* b2 = (const float*)d_in[28];

  const int* src = eidx;          // edge_index[0]
  const int* dst = eidx + NE;     // edge_index[1]

  // workspace layout (floats)
  float* ws = (float*)d_ws;
  float* hA      = ws;                       // N*64
  float* hB      = hA + (size_t)NN * DD;     // N*64
  float* agg     = hB + (size_t)NN * DD;     // N*64
  float* inv_cnt = agg + (size_t)NN * DD;    // N
  float* gsum    = inv_cnt + NN;             // G*64
  float* gcnt    = gsum + GG * DD;           // G

  const int ZB = 256;

  // degrees (once; graph is static across layers)
  zero_f32<<<(NN + ZB - 1) / ZB, ZB, 0, stream>>>(inv_cnt, NN);
  count_edges<<<(NE + ZB - 1) / ZB, ZB, 0, stream>>>(dst, inv_cnt);
  invert_cnt<<<(NN + ZB - 1) / ZB, ZB, 0, stream>>>(inv_cnt);

  const int aggN       = NN * DD;
  const int aggBlocks  = 2048;                                  // grid-stride zero
  const int edgeBlocks = (NE * 32 + ZB - 1) / ZB;               // wave per edge
  const int tileBlocks = ((NN + 15) / 16 + 7) / 8;              // wave per 16-row tile

  const float* hin = x;
  float* layer_out[3] = {hA, hB, hA};
  for (int l = 0; l < 3; ++l) {
    zero_f32<<<aggBlocks, ZB, 0, stream>>>(agg, aggN);
    edge_aggregate<<<edgeBlocks, ZB, 0, stream>>>(hin, src, dst, agg);
    sage_layer_wmma<<<tileBlocks, ZB, 0, stream>>>(
        hin, agg, inv_cnt,
        P[l][0], P[l][1], P[l][2], P[l][3], P[l][4], P[l][5], P[l][6],
        layer_out[l]);
    hin = layer_out[l];
  }

  // global mean pool + head
  zero_f32<<<(GG * DD + GG + ZB - 1) / ZB, ZB, 0, stream>>>(gsum, GG * DD + GG);
  const int poolWaves  = (NN + POOL_CHUNK - 1) / POOL_CHUNK;
  const int poolBlocks = (poolWaves * 32 + ZB - 1) / ZB;
  pool_nodes<<<poolBlocks, ZB, 0, stream>>>(hin, batch, gsum, gcnt);
  classifier<<<1, 128, 0, stream>>>(gsum, gcnt, gattr, W1, b1, W2, b2, (float*)d_out);
}